// AdaptiveMixGNN_10213432229977
// MI455X (gfx1250) — compile-verified
//
#include <hip/hip_runtime.h>
#include <hip/hip_bf16.h>

typedef __attribute__((ext_vector_type(2))) float v2f;
typedef __attribute__((ext_vector_type(8))) float v8f;

#define FDIM 128
#define SCAN_THREADS 1024

// ---------------- CSR build: histogram ----------------
__global__ void count_init_kernel(int* cnt, int n) {
    int i = blockIdx.x * blockDim.x + threadIdx.x;
    if (i < n) cnt[i] = 0;
}

__global__ void count_accum_kernel(const int* __restrict__ row, int* cnt, int E) {
    int e = blockIdx.x * blockDim.x + threadIdx.x;
    if (e < E) atomicAdd(&cnt[row[e]], 1);
}

// dinv[i] = rsqrt(deg) with deg = cnt + 1 (self loop); matches D^-1/2 of ref.
__global__ void dinv_kernel(const int* __restrict__ cnt, float* dinv, int n) {
    int i = blockIdx.x * blockDim.x + threadIdx.x;
    if (i < n) {
        float d = (float)(cnt[i] + 1);
        dinv[i] = rsqrtf(d);
    }
}

// ---------------- single-block exclusive scan (N up to ~any, chunked) ----------------
__global__ void scan_kernel(const int* __restrict__ cnt, int* rowptr, int n) {
    __shared__ int sdata[SCAN_THREADS];
    __shared__ int carry;
    int tid = threadIdx.x;
    if (tid == 0) carry = 0;
    __syncthreads();
    for (int base = 0; base < n; base += SCAN_THREADS) {
        int i = base + tid;
        int v = (i < n) ? cnt[i] : 0;
        sdata[tid] = v;
        __syncthreads();
        // Hillis-Steele inclusive scan
        for (int off = 1; off < SCAN_THREADS; off <<= 1) {
            int t = (tid >= off) ? sdata[tid - off] : 0;
            __syncthreads();
            sdata[tid] += t;
            __syncthreads();
        }
        if (i < n) rowptr[i + 1] = carry + sdata[tid];   // inclusive -> rowptr[i+1]
        __syncthreads();
        if (tid == SCAN_THREADS - 1) carry += sdata[SCAN_THREADS - 1];
        __syncthreads();
    }
    if (tid == 0) rowptr[0] = 0;
}

__global__ void cursor_copy_kernel(const int* __restrict__ rowptr, int* cursor, int n) {
    int i = blockIdx.x * blockDim.x + threadIdx.x;
    if (i < n) cursor[i] = rowptr[i];
}

// ---------------- CSR fill: per-edge, store col + precomputed weight ----------------
__global__ void csr_fill_kernel(const int* __restrict__ row, const int* __restrict__ col,
                                const float* __restrict__ dinv,
                                int* cursor, int* __restrict__ ccol, float* __restrict__ cw,
                                int E) {
    int e = blockIdx.x * blockDim.x + threadIdx.x;
    if (e < E) {
        int r = row[e];
        int c = col[e];
        int pos = atomicAdd(&cursor[r], 1);
        ccol[pos] = c;
        cw[pos]   = dinv[r] * dinv[c];
    }
}

// ---------------- fused sparse layer: gather-aggregate + alpha + mix ----------------
// One 128-thread block per node i, lane f owns feature f:
//   zlp_f = sum_{j in row i} w_j * X[col_j][f]  (register accumulation, no atomics)
//   alpha = sigmoid(dot(X[i], aw) + ab)         (wave32 shuffle + LDS reduce)
//   z[i][f] = (1-alpha)*x + (2*alpha-1)*(zlp + dinv^2 * x)
__global__ void fused_sparse_kernel(const float* __restrict__ X,
                                    const int* __restrict__ rowptr,
                                    const int* __restrict__ ccol,
                                    const float* __restrict__ cw,
                                    const float* __restrict__ dinv,
                                    const float* __restrict__ aw,
                                    const float* __restrict__ ab,
                                    float* __restrict__ Z, int n) {
    __shared__ float red[4];
    int i = blockIdx.x;
    int t = threadIdx.x;
    size_t base = (size_t)i * FDIM + t;
    float xv = X[base];

    // alpha = sigmoid(x . aw + ab)
    float p = xv * aw[t];
    #pragma unroll
    for (int off = 16; off > 0; off >>= 1) p += __shfl_down(p, off, 32);
    if ((t & 31) == 0) red[t >> 5] = p;
    __syncthreads();
    if (t == 0) {
        float s = red[0] + red[1] + red[2] + red[3] + ab[0];
        red[0] = 1.0f / (1.0f + __expf(-s));
    }

    // gather aggregation while the reduction settles
    int start = rowptr[i];
    int end   = rowptr[i + 1];
    float acc = 0.0f;
    for (int j = start; j < end; ++j) {
        int   c = ccol[j];   // block-uniform load
        float w = cw[j];     // block-uniform load
        acc += w * X[(size_t)c * FDIM + t];   // coalesced row gather
    }

    __syncthreads();
    float alpha = red[0];
    float di = dinv[i];
    float zlp = acc + di * di * xv;           // self-loop term
    Z[base] = (1.0f - alpha) * xv + (2.0f * alpha - 1.0f) * zlp;
}

// ---------------- dense projection via fp32 WMMA ----------------
// Out[n][o] = sum_k Z[n][k] * W[o][k] + bias[o]   (torch layout W[out][in])
// One wave computes a 16x16 output tile with 32 chained 16x16x4 f32 WMMAs.
__global__ void gemm_wmma_kernel(const float* __restrict__ Z,
                                 const float* __restrict__ W,
                                 const float* __restrict__ bias,
                                 float* __restrict__ Out,
                                 int nNodes, int Cout, int relu) {
    int wave = threadIdx.x >> 5;     // output-channel tile index
    int lane = threadIdx.x & 31;
    int half = lane >> 4;            // 0: K sub {0,1}, 1: K sub {2,3}
    int l    = lane & 15;

    int n0 = blockIdx.x * 16;
    int o0 = wave * 16;

    // Clamp row reads so EXEC stays all-ones through the WMMAs; stores guarded.
    int zr = n0 + l; if (zr > nNodes - 1) zr = nNodes - 1;
    int wr = o0 + l; if (wr > Cout   - 1) wr = Cout   - 1;
    const float* zrow = Z + (size_t)zr * FDIM;
    const float* wrow = W + (size_t)wr * FDIM;

    v8f acc = {0.f, 0.f, 0.f, 0.f, 0.f, 0.f, 0.f, 0.f};
    #pragma unroll
    for (int k = 0; k < FDIM; k += 4) {
        v2f a, b;
        a.x = zrow[k + half * 2 + 0];
        a.y = zrow[k + half * 2 + 1];
        b.x = wrow[k + half * 2 + 0];
        b.y = wrow[k + half * 2 + 1];
        acc = __builtin_amdgcn_wmma_f32_16x16x4_f32(
            /*neg_a=*/false, a, /*neg_b=*/false, b,
            /*c_mod=*/(short)0, acc, /*reuse_a=*/false, /*reuse_b=*/false);
    }

    // C/D layout: VGPR r, lanes 0-15 -> M=r, lanes 16-31 -> M=r+8; N=l
    int o = o0 + l;
    if (o < Cout) {
        float bv = bias[o];
        #pragma unroll
        for (int r = 0; r < 8; ++r) {
            int node = n0 + half * 8 + r;
            if (node < nNodes) {
                float v = acc[r] + bv;
                if (relu) v = v > 0.0f ? v : 0.0f;
                Out[(size_t)node * Cout + o] = v;
            }
        }
    }
}

extern "C" void kernel_launch(void* const* d_in, const int* in_sizes, int n_in,
                              void* d_out, int out_size, void* d_ws, size_t ws_size,
                              hipStream_t stream) {
    const float* x   = (const float*)d_in[0];
    const int*   ei  = (const int*)  d_in[1];   // [2, E]: row then col
    const float* aw1 = (const float*)d_in[2];
    const float* ab1 = (const float*)d_in[3];
    const float* W1  = (const float*)d_in[4];
    const float* b1  = (const float*)d_in[5];
    const float* aw2 = (const float*)d_in[6];
    const float* ab2 = (const float*)d_in[7];
    const float* W2  = (const float*)d_in[8];
    const float* b2  = (const float*)d_in[9];

    const int N = in_sizes[0] / FDIM;
    const int E = in_sizes[1] / 2;
    const int H = FDIM;
    const int C = 40;

    const int* row = ei;
    const int* col = ei + E;

    // workspace carve-up (bytes)
    char* ws = (char*)d_ws;
    float* dinv   = (float*)ws;                          ws += (size_t)N * 4;
    float* zbuf   = (float*)ws;                          ws += (size_t)N * H * 4;
    float* hbuf   = (float*)ws;                          ws += (size_t)N * H * 4;
    int*   cnt    = (int*)ws;                            ws += (size_t)N * 4;
    int*   rowptr = (int*)ws;                            ws += (size_t)(N + 1) * 4;
    int*   cursor = (int*)ws;                            ws += (size_t)N * 4;
    int*   ccol   = (int*)ws;                            ws += (size_t)E * 4;
    float* cw     = (float*)ws;                          ws += (size_t)E * 4;

    float* outp = (float*)d_out;

    int nb  = (N + 255) / 256;
    int eb  = (E + 255) / 256;
    int ntiles = (N + 15) / 16;

    // --- CSR build (graph shared by both layers) ---
    count_init_kernel<<<nb, 256, 0, stream>>>(cnt, N);
    count_accum_kernel<<<eb, 256, 0, stream>>>(row, cnt, E);
    dinv_kernel<<<nb, 256, 0, stream>>>(cnt, dinv, N);
    scan_kernel<<<1, SCAN_THREADS, 0, stream>>>(cnt, rowptr, N);
    cursor_copy_kernel<<<nb, 256, 0, stream>>>(rowptr, cursor, N);
    csr_fill_kernel<<<eb, 256, 0, stream>>>(row, col, dinv, cursor, ccol, cw, E);

    // --- layer 1: x -> hbuf (relu) ---
    fused_sparse_kernel<<<N, FDIM, 0, stream>>>(x, rowptr, ccol, cw, dinv, aw1, ab1, zbuf, N);
    gemm_wmma_kernel<<<ntiles, (H / 16) * 32, 0, stream>>>(zbuf, W1, b1, hbuf, N, H, 1);

    // --- layer 2: hbuf -> out (no relu) ---
    fused_sparse_kernel<<<N, FDIM, 0, stream>>>(hbuf, rowptr, ccol, cw, dinv, aw2, ab2, zbuf, N);
    gemm_wmma_kernel<<<ntiles, ((C + 15) / 16) * 32, 0, stream>>>(zbuf, W2, b2, outp, N, C, 0);
}